// PauliStringOperator_26594437496881
// MI455X (gfx1250) — compile-verified
//
#include <hip/hip_runtime.h>

// PauliStringOperator on a 23-qubit state, N = 2^23 amplitudes, batch B = 2.
//
//   XOR permutation mask = bits {22,19,15,9} = 0x488200
//   phase[i] is REAL ±1:  +1 if parity(bits 19,17,12,9 of i) is odd, else -1
//   out[b,c,i] = 0.5 * phase[i] * state_c[b, i ^ mask]
//
// Pure streaming kernel: 256 MiB traffic, 0 FLOPs -> bandwidth bound
// (~11.5 us floor at 23.3 TB/s). Mask's lowest bit is 9, so 8-element
// (32 B) aligned output groups gather contiguous aligned source groups and
// share one sign: fully coalesced b128 on both sides.
//
// L2 strategy (192 MB global L2): the 128 MiB input set fits -> regular RT
// loads keep it L2-resident across graph replays; the 128 MiB output is
// written with non-temporal b128 stores (th:NT) so the write stream does not
// evict the inputs.
//
// All eight plane base pointers are passed from the host so every access is
// saddr-form (scalar 64-bit base + one shared 32-bit lane offset + imm).

typedef float v4f __attribute__((ext_vector_type(4)));

namespace {
constexpr unsigned kXorMask = 0x488200u;  // bits 22, 19, 15, 9
}

__global__ __launch_bounds__(256) void pauli_apply_kernel(
    const float* __restrict__ src0,   // b=0 re  (state_re + 0)
    const float* __restrict__ src1,   // b=0 im  (state_im + 0)
    const float* __restrict__ src2,   // b=1 re  (state_re + N)
    const float* __restrict__ src3,   // b=1 im  (state_im + N)
    float* __restrict__ dst0,         // out + 0*N
    float* __restrict__ dst1,         // out + 1*N
    float* __restrict__ dst2,         // out + 2*N
    float* __restrict__ dst3)         // out + 3*N
{
  // Each thread: 8 consecutive output elements (two float4 slots) per plane.
  const unsigned i = (blockIdx.x * 256u + threadIdx.x) * 8u;  // output index
  const unsigned j = i ^ kXorMask;                            // source index

  // Sign: +0.5 when parity of bits {19,17,12,9} is odd, else -0.5.
  // All parity bits >= 9, so one sign covers the whole 8-element group.
  const unsigned p = (i >> 19) ^ (i >> 17) ^ (i >> 12) ^ (i >> 9);
  const float s = (p & 1u) ? 0.5f : -0.5f;

  // Gather two float4 slots from each plane (saddr b128, imm offset +16
  // for the second slot of each plane).
  const v4f a0 = *reinterpret_cast<const v4f*>(src0 + j);
  const v4f b0 = *reinterpret_cast<const v4f*>(src0 + j + 4u);
  const v4f a1 = *reinterpret_cast<const v4f*>(src1 + j);
  const v4f b1 = *reinterpret_cast<const v4f*>(src1 + j + 4u);
  const v4f a2 = *reinterpret_cast<const v4f*>(src2 + j);
  const v4f b2 = *reinterpret_cast<const v4f*>(src2 + j + 4u);
  const v4f a3 = *reinterpret_cast<const v4f*>(src3 + j);
  const v4f b3 = *reinterpret_cast<const v4f*>(src3 + j + 4u);

  // Streaming non-temporal b128 stores (gfx1250 th:NT cache policy).
  __builtin_nontemporal_store(a0 * s, reinterpret_cast<v4f*>(dst0 + i));
  __builtin_nontemporal_store(b0 * s, reinterpret_cast<v4f*>(dst0 + i + 4u));
  __builtin_nontemporal_store(a1 * s, reinterpret_cast<v4f*>(dst1 + i));
  __builtin_nontemporal_store(b1 * s, reinterpret_cast<v4f*>(dst1 + i + 4u));
  __builtin_nontemporal_store(a2 * s, reinterpret_cast<v4f*>(dst2 + i));
  __builtin_nontemporal_store(b2 * s, reinterpret_cast<v4f*>(dst2 + i + 4u));
  __builtin_nontemporal_store(a3 * s, reinterpret_cast<v4f*>(dst3 + i));
  __builtin_nontemporal_store(b3 * s, reinterpret_cast<v4f*>(dst3 + i + 4u));
}

extern "C" void kernel_launch(void* const* d_in, const int* in_sizes, int n_in,
                              void* d_out, int out_size, void* d_ws, size_t ws_size,
                              hipStream_t stream) {
  const float* state_re = reinterpret_cast<const float*>(d_in[0]);  // [2, N] f32
  const float* state_im = reinterpret_cast<const float*>(d_in[1]);  // [2, N] f32
  float* out = reinterpret_cast<float*>(d_out);                     // [2, 2, N] f32

  const unsigned n = static_cast<unsigned>(in_sizes[0] / 2);  // N = 8388608 = 2^23
  // 8 elements per thread, 256 threads per block -> 2048 elements per block.
  // N = 2^23 is exactly divisible by 2048 -> no tail guard needed.
  const unsigned blocks = n / (8u * 256u);                    // 4096 blocks

  pauli_apply_kernel<<<dim3(blocks), dim3(256), 0, stream>>>(
      state_re,          // b=0 re
      state_im,          // b=0 im
      state_re + n,      // b=1 re
      state_im + n,      // b=1 im
      out + 0u * n,      // [0,0,:]
      out + 1u * n,      // [0,1,:]
      out + 2u * n,      // [1,0,:]
      out + 3u * n);     // [1,1,:]
}